// InsNegLoss_35905926594960
// MI455X (gfx1250) — compile-verified
//
#include <hip/hip_runtime.h>

typedef __attribute__((ext_vector_type(2))) float v2f;
typedef __attribute__((ext_vector_type(8))) float v8f;

constexpr int N  = 8192;
constexpr int D  = 256;
constexpr int KC = D / 4;                     // 64 k-chunks of 4 (one WMMA each)
constexpr int CSPLIT = 4;                     // column splits (partial stats)
constexpr int COLS_PER_SPLIT = N / CSPLIT;    // 2048
constexpr int CTILES = COLS_PER_SPLIT / 16;   // 128 column tiles per block
constexpr int WPB = 4;                        // waves per block
constexpr int ROWS_PER_BLOCK = WPB * 16;      // 64
constexpr int LDS_STRIDE = 260;               // padded floats per staged column (bank-conflict free)

// Phase 1: fused GEMM + per-row streaming stats (sum, max, tie-count).
__global__ __launch_bounds__(128) void simstats_kernel(
    const float* __restrict__ zi, const float* __restrict__ zj,
    float* __restrict__ posw, float* __restrict__ sumw, int* __restrict__ cntw)
{
  __shared__ __align__(16) float buf[2][16 * LDS_STRIDE];

  const int tid  = threadIdx.x;
  const int lane = tid & 31;
  const int wave = tid >> 5;
  const int r    = lane & 15;   // row (A) / column (B) index within tile
  const int hi   = lane >> 4;   // K-pair select

  const int p       = blockIdx.x & (CSPLIT - 1);
  const int rb      = blockIdx.x >> 2;
  const int rowbase = rb * ROWS_PER_BLOCK + wave * 16;
  const int colstart = p * COLS_PER_SPLIT;

  // ---- preload this wave's 16xD A-panel into registers (WMMA A layout) ----
  const float* arow = zi + (size_t)(rowbase + r) * D + 2 * hi;
  v2f a[KC];
#pragma unroll
  for (int c = 0; c < KC; ++c)
    a[c] = *reinterpret_cast<const v2f*>(arow + 4 * c);

  float sum[8], mx[8];
  int   cnt[8];
#pragma unroll
  for (int v = 0; v < 8; ++v) { sum[v] = 0.f; mx[v] = -3.402823466e38f; cnt[v] = 0; }

  // ---- prologue: cooperatively stage B tile 0 into LDS ----
  float4 pre[8];
#pragma unroll
  for (int i = 0; i < 8; ++i) {
    int idx = i * 128 + tid, col = idx >> 6, w = idx & 63;
    pre[i] = *reinterpret_cast<const float4*>(zj + (size_t)(colstart + col) * D + 4 * w);
  }
#pragma unroll
  for (int i = 0; i < 8; ++i) {
    int idx = i * 128 + tid, col = idx >> 6, w = idx & 63;
    *reinterpret_cast<float4*>(&buf[0][col * LDS_STRIDE + 4 * w]) = pre[i];
  }
  __syncthreads();

  for (int jt = 0; jt < CTILES; ++jt) {
    const int cur = jt & 1;

    // prefetch next B tile into registers (overlaps the WMMA chain below)
    if (jt + 1 < CTILES) {
      const float* src = zj + (size_t)(colstart + (jt + 1) * 16) * D;
#pragma unroll
      for (int i = 0; i < 8; ++i) {
        int idx = i * 128 + tid, col = idx >> 6, w = idx & 63;
        pre[i] = *reinterpret_cast<const float4*>(src + (size_t)col * D + 4 * w);
      }
    }

    // ---- 16x16 output tile: 64 chained V_WMMA_F32_16X16X4_F32 ----
    v8f acc = {0.f, 0.f, 0.f, 0.f, 0.f, 0.f, 0.f, 0.f};
    const float* bp = &buf[cur][r * LDS_STRIDE + 2 * hi];
#pragma unroll
    for (int c = 0; c < KC; ++c) {
      float2 t2 = *reinterpret_cast<const float2*>(bp + 4 * c);
      v2f b; b.x = t2.x; b.y = t2.y;
      acc = __builtin_amdgcn_wmma_f32_16x16x4_f32(false, a[c], false, b,
                                                  (short)0, acc, false, false);
    }

    // ---- per-lane streaming row stats (max with exact tie counting) ----
#pragma unroll
    for (int v = 0; v < 8; ++v) {
      float x = acc[v];
      sum[v] += x;
      if (x > mx[v])       { mx[v] = x; cnt[v] = 1; }
      else if (x == mx[v]) { cnt[v] += 1; }
    }

    // stage next tile into the other LDS buffer, then hand off
    if (jt + 1 < CTILES) {
      float* dst = &buf[cur ^ 1][0];
#pragma unroll
      for (int i = 0; i < 8; ++i) {
        int idx = i * 128 + tid, col = idx >> 6, w = idx & 63;
        *reinterpret_cast<float4*>(&dst[col * LDS_STRIDE + 4 * w]) = pre[i];
      }
    }
    __syncthreads();
  }

  // ---- combine the 16 column-lanes within each half-wave ----
#pragma unroll
  for (int v = 0; v < 8; ++v) {
#pragma unroll
    for (int off = 8; off >= 1; off >>= 1) {
      float omx = __shfl_xor(mx[v],  off);
      float osm = __shfl_xor(sum[v], off);
      int   ocn = __shfl_xor(cnt[v], off);
      sum[v] += osm;
      if (omx > mx[v])       { mx[v] = omx; cnt[v] = ocn; }
      else if (omx == mx[v]) { cnt[v] += ocn; }
    }
  }

  // lanes 0 and 16 hold final stats for rows rowbase+0..7 / +8..15
  if (r == 0) {
    const int rowo = rowbase + hi * 8;
#pragma unroll
    for (int v = 0; v < 8; ++v) {
      const int row = rowo + v;
      posw[p * N + row] = mx[v];
      sumw[p * N + row] = sum[v];
      cntw[p * N + row] = cnt[v];
    }
  }
}

// Phase 2: combine column-split partials, reduce max_neg, emit scalar loss.
__global__ __launch_bounds__(1024) void finalize_kernel(
    const float* __restrict__ posw, const float* __restrict__ sumw,
    const int* __restrict__ cntw, float* __restrict__ out)
{
  __shared__ double s1[1024];
  __shared__ double s2[1024];
  __shared__ int    sm[1024];
  const int t = threadIdx.x;

  float posr[8], nsr[8];
  int lm = 0;
#pragma unroll
  for (int i = 0; i < 8; ++i) {
    const int row = t + i * 1024;
    float mxv = -3.402823466e38f; int c = 0; float s = 0.f;
#pragma unroll
    for (int p = 0; p < CSPLIT; ++p) {
      float m  = posw[p * N + row];
      int   cc = cntw[p * N + row];
      s += sumw[p * N + row];
      if (m > mxv)       { mxv = m; c = cc; }
      else if (m == mxv) { c += cc; }
    }
    posr[i] = mxv;
    nsr[i]  = s - mxv * (float)c;        // sum of strictly-below-max entries
    int neg = N - c;
    lm = neg > lm ? neg : lm;
  }
  sm[t] = lm;
  __syncthreads();
  for (int sft = 512; sft > 0; sft >>= 1) {
    if (t < sft) sm[t] = sm[t] > sm[t + sft] ? sm[t] : sm[t + sft];
    __syncthreads();
  }
  const float maxneg = (float)sm[0];

  double l1 = 0.0, l2 = 0.0;
#pragma unroll
  for (int i = 0; i < 8; ++i) {
    float pos = posr[i], ns = nsr[i];
    l1 += (double)(fminf(ns, 30.f) - pos);          // neg_sum_c - pos
    float tr = pos - ns / maxneg + 1.f;             // triplet pre-relu
    l2 += (double)fmaxf(tr, 0.f);
  }
  s1[t] = l1; s2[t] = l2;
  __syncthreads();
  for (int sft = 512; sft > 0; sft >>= 1) {
    if (t < sft) { s1[t] += s1[t + sft]; s2[t] += s2[t + sft]; }
    __syncthreads();
  }
  if (t == 0) out[0] = (float)((s1[0] + s2[0]) / (double)N);
}

extern "C" void kernel_launch(void* const* d_in, const int* in_sizes, int n_in,
                              void* d_out, int out_size, void* d_ws, size_t ws_size,
                              hipStream_t stream) {
  const float* zi = (const float*)d_in[0];
  const float* zj = (const float*)d_in[1];

  float* posw = (float*)d_ws;                 // CSPLIT * N floats
  float* sumw = posw + CSPLIT * N;            // CSPLIT * N floats
  int*   cntw = (int*)(sumw + CSPLIT * N);    // CSPLIT * N ints  (384 KB total)

  simstats_kernel<<<(N / ROWS_PER_BLOCK) * CSPLIT, 128, 0, stream>>>(zi, zj, posw, sumw, cntw);
  finalize_kernel<<<1, 1024, 0, stream>>>(posw, sumw, cntw, (float*)d_out);
}